// GATConv_43671227466143
// MI455X (gfx1250) — compile-verified
//
#include <hip/hip_runtime.h>
#include <hip/hip_bf16.h>

#define IN_DIM   256
#define OUTD     16
#define HEADS    8
#define FDIM     128            // OUTD * HEADS
#define NEG_SLOPE 0.2f

typedef float v2f __attribute__((ext_vector_type(2)));
typedef float v8f __attribute__((ext_vector_type(8)));

// ---- monotone float <-> ordered-int mapping (involution) ----
__device__ __forceinline__ int f2ord(float f) {
    int i = __float_as_int(f);
    return i >= 0 ? i : (i ^ 0x7fffffff);
}
__device__ __forceinline__ float ord2f(int i) {
    return __int_as_float(i >= 0 ? i : (i ^ 0x7fffffff));
}

// =====================================================================
// Kernel 1: Wh = h @ W  via V_WMMA_F32_16X16X4_F32
// block = 256 threads = 8 waves; wave w computes the 16x16 tile for head w
// grid.x = ceil(N/16) row tiles
// =====================================================================
__global__ __launch_bounds__(256) void gat_gemm_wmma(
    const float* __restrict__ h, const float* __restrict__ W,
    float* __restrict__ Wh, int N)
{
    const int wave    = threadIdx.x >> 5;
    const int lane    = threadIdx.x & 31;
    const int rowBase = blockIdx.x * 16;
    const int colBase = wave * 16;          // head tile
    const int mrow    = lane & 15;          // A-matrix row this lane feeds
    const int khalf   = lane >> 4;          // 0: K=0,1 ; 1: K=2,3
    const int ncol    = lane & 15;          // B/C/D column this lane holds

    int arow = rowBase + mrow;
    if (arow >= N) arow = N - 1;            // branchless clamp (EXEC stays full)
    const float* __restrict__ hrow = h + (size_t)arow * IN_DIM;

    v8f acc = {};
    #pragma unroll 4
    for (int k = 0; k < IN_DIM; k += 4) {
        const int ka = k + khalf * 2;
        v2f a, b;
        a.x = hrow[ka + 0];
        a.y = hrow[ka + 1];
        b.x = W[(size_t)(ka + 0) * FDIM + colBase + ncol];
        b.y = W[(size_t)(ka + 1) * FDIM + colBase + ncol];
        acc = __builtin_amdgcn_wmma_f32_16x16x4_f32(
            /*neg_a=*/false, a, /*neg_b=*/false, b,
            /*c_mod=*/(short)0, acc, /*reuse_a=*/false, /*reuse_b=*/false);
    }

    // C/D layout: VGPR r, lanes 0-15 -> M=r ; lanes 16-31 -> M=r+8
    // Block-uniform bounds test: a 16-row tile is entirely in or out.
    float* __restrict__ dst =
        Wh + (size_t)(rowBase + khalf * 8) * FDIM + colBase + ncol;
    if (rowBase + 16 <= N) {                // uniform fast path: straight-line stores
        #pragma unroll
        for (int r = 0; r < 8; ++r)
            dst[(size_t)r * FDIM] = acc[r];
    } else {                                 // rare ragged tile
        #pragma unroll
        for (int r = 0; r < 8; ++r) {
            const int m = rowBase + r + khalf * 8;
            if (m < N) Wh[(size_t)m * FDIM + colBase + ncol] = acc[r];
        }
    }
}

// =====================================================================
// Kernel 2: per-(node,head) attention scores Wh1 = <Wh, a_l>, Wh2 = <Wh, a_r>
// =====================================================================
__global__ __launch_bounds__(256) void gat_scores(
    const float* __restrict__ Wh, const float* __restrict__ a_l,
    const float* __restrict__ a_r, float* __restrict__ Wh1,
    float* __restrict__ Wh2, int N)
{
    const int t = blockIdx.x * blockDim.x + threadIdx.x;
    if (t >= N * HEADS) return;
    const int n  = t >> 3;
    const int hd = t & 7;
    const float* __restrict__ p  = Wh + (size_t)n * FDIM + hd * OUTD;
    const float* __restrict__ al = a_l + hd * OUTD;
    const float* __restrict__ ar = a_r + hd * OUTD;
    float s1 = 0.f, s2 = 0.f;
    #pragma unroll
    for (int d = 0; d < OUTD; ++d) {
        const float v = p[d];
        s1 = fmaf(v, al[d], s1);
        s2 = fmaf(v, ar[d], s2);
    }
    Wh1[t] = s1;
    Wh2[t] = s2;
}

// =====================================================================
// Kernel 3: init accumulators (every call: graph replays don't re-poison)
// =====================================================================
__global__ __launch_bounds__(256) void gat_init(
    int* __restrict__ m_int, float* __restrict__ s,
    float* __restrict__ out, int N)
{
    const int t = blockIdx.x * blockDim.x + threadIdx.x;
    if (t < N * FDIM) out[t] = 0.f;
    if (t < N * HEADS) { m_int[t] = 0x80000000; s[t] = 0.f; }
}

// =====================================================================
// Kernel 4: segment max of leaky-relu'd edge logits (exact via ordered-int)
// one thread per (edge, head)
// =====================================================================
__global__ __launch_bounds__(256) void gat_edge_max(
    const int* __restrict__ row, const int* __restrict__ col,
    const float* __restrict__ Wh1, const float* __restrict__ Wh2,
    int* __restrict__ m_int, int E)
{
    const int t = blockIdx.x * blockDim.x + threadIdx.x;
    if (t >= E * HEADS) return;
    const int e  = t >> 3;
    const int hd = t & 7;
    const int r  = row[e];
    const int c  = col[e];
    float ev = Wh1[r * HEADS + hd] + Wh2[c * HEADS + hd];
    ev = ev > 0.f ? ev : NEG_SLOPE * ev;
    atomicMax(&m_int[r * HEADS + hd], f2ord(ev));
}

// =====================================================================
// Kernel 5: segment sum of exp(e - m)
// =====================================================================
__global__ __launch_bounds__(256) void gat_edge_sum(
    const int* __restrict__ row, const int* __restrict__ col,
    const float* __restrict__ Wh1, const float* __restrict__ Wh2,
    const int* __restrict__ m_int, float* __restrict__ s, int E)
{
    const int t = blockIdx.x * blockDim.x + threadIdx.x;
    if (t >= E * HEADS) return;
    const int e  = t >> 3;
    const int hd = t & 7;
    const int r  = row[e];
    const int c  = col[e];
    float ev = Wh1[r * HEADS + hd] + Wh2[c * HEADS + hd];
    ev = ev > 0.f ? ev : NEG_SLOPE * ev;
    const float mf = ord2f(m_int[r * HEADS + hd]);
    atomicAdd(&s[r * HEADS + hd], __expf(ev - mf));
}

// =====================================================================
// Kernel 6: aggregation. One wave per edge; lane l handles the 4
// contiguous elements [4l, 4l+4) of the 128-float feature row.
// out[r, d, h] += alpha[e, h] * Wh[c, d*8 + h]   (flat index i = d*8+h)
// =====================================================================
__global__ __launch_bounds__(256) void gat_aggregate(
    const int* __restrict__ row, const int* __restrict__ col,
    const float* __restrict__ Wh, const float* __restrict__ Wh1,
    const float* __restrict__ Wh2, const int* __restrict__ m_int,
    const float* __restrict__ s, float* __restrict__ out, int E)
{
    const int e = blockIdx.x * 8 + (threadIdx.x >> 5);
    if (e >= E) return;
    const int lane = threadIdx.x & 31;
    const int r = row[e];
    const int c = col[e];

    // every lane computes alpha for head (lane & 7); lanes 8..31 are replicas
    const int hd = lane & 7;
    float ev = Wh1[r * HEADS + hd] + Wh2[c * HEADS + hd];
    ev = ev > 0.f ? ev : NEG_SLOPE * ev;
    const float mf = ord2f(m_int[r * HEADS + hd]);
    const float alpha = __expf(ev - mf) / s[r * HEADS + hd];

    const int base = lane * 4;                       // flat feature index
    const float4 v = ((const float4*)(Wh + (size_t)c * FDIM))[lane];

    float4 w;
    w.x = v.x * __shfl(alpha, (base + 0) & 7, 32);
    w.y = v.y * __shfl(alpha, (base + 1) & 7, 32);
    w.z = v.z * __shfl(alpha, (base + 2) & 7, 32);
    w.w = v.w * __shfl(alpha, (base + 3) & 7, 32);

    float* dst = out + (size_t)r * FDIM + base;
    atomicAdd(dst + 0, w.x);
    atomicAdd(dst + 1, w.y);
    atomicAdd(dst + 2, w.z);
    atomicAdd(dst + 3, w.w);
}

// =====================================================================
// Kernel 7: in-place ReLU on the accumulated output
// =====================================================================
__global__ __launch_bounds__(256) void gat_relu(float* __restrict__ out, int n)
{
    const int t = blockIdx.x * blockDim.x + threadIdx.x;
    if (t < n) out[t] = fmaxf(out[t], 0.f);
}

// =====================================================================
extern "C" void kernel_launch(void* const* d_in, const int* in_sizes, int n_in,
                              void* d_out, int out_size, void* d_ws, size_t ws_size,
                              hipStream_t stream)
{
    const int*   row = (const int*)d_in[0];
    const int*   col = (const int*)d_in[1];
    const float* h   = (const float*)d_in[2];
    const float* W   = (const float*)d_in[3];
    const float* a_l = (const float*)d_in[4];
    const float* a_r = (const float*)d_in[5];
    float*       out = (float*)d_out;

    const int E = in_sizes[0];
    const int N = in_sizes[2] / IN_DIM;

    // workspace layout
    char* ws = (char*)d_ws;
    float* Wh    = (float*)ws;                 ws += (size_t)N * FDIM  * sizeof(float);
    float* Wh1   = (float*)ws;                 ws += (size_t)N * HEADS * sizeof(float);
    float* Wh2   = (float*)ws;                 ws += (size_t)N * HEADS * sizeof(float);
    int*   m_int = (int*)ws;                   ws += (size_t)N * HEADS * sizeof(int);
    float* s     = (float*)ws;                 ws += (size_t)N * HEADS * sizeof(float);

    const int B = 256;

    // 1) GEMM (WMMA f32)
    gat_gemm_wmma<<<(N + 15) / 16, B, 0, stream>>>(h, W, Wh, N);

    // 2) per-(node,head) scores
    gat_scores<<<(N * HEADS + B - 1) / B, B, 0, stream>>>(Wh, a_l, a_r, Wh1, Wh2, N);

    // 3) init m / s / out (must re-init every call for graph replay)
    gat_init<<<(N * FDIM + B - 1) / B, B, 0, stream>>>(m_int, s, out, N);

    // 4) segment max
    gat_edge_max<<<((size_t)E * HEADS + B - 1) / B, B, 0, stream>>>(row, col, Wh1, Wh2, m_int, E);

    // 5) segment sum of exp
    gat_edge_sum<<<((size_t)E * HEADS + B - 1) / B, B, 0, stream>>>(row, col, Wh1, Wh2, m_int, s, E);

    // 6) weighted aggregation (one wave per edge)
    gat_aggregate<<<(E + 7) / 8, B, 0, stream>>>(row, col, Wh, Wh1, Wh2, m_int, s, out, E);

    // 7) final ReLU
    gat_relu<<<(N * FDIM + B - 1) / B, B, 0, stream>>>(out, N * FDIM);
}